// JointNetwork_30176440222156
// MI455X (gfx1250) — compile-verified
//
#include <hip/hip_runtime.h>
#include <hip/hip_bf16.h>

// Problem constants (from reference)
#define N_  32
#define T_  76
#define U_  20
#define H_  2
#define C_  4096
#define F_  (T_*U_*H_)      // 3040
#define ENC_IN_  38912
#define PRED_IN_ 15872
#define NEGC (-1e9f)

#define GEMM_WAVES 8        // K-split factor; Kin/GEMM_WAVES must be a multiple of 32

typedef float v2f __attribute__((ext_vector_type(2)));
typedef float v8f __attribute__((ext_vector_type(8)));

// ---------------------------------------------------------------------------
// Kernel 1: Y(32 x Fdim) = X(32 x Kin) @ W(Fdim x Kin)^T + bias
//
// grid.x = Fdim/16 tiles. Each block = 8 waves; wave w covers K-chunk
// [w*Kc, (w+1)*Kc) with v_wmma_f32_16x16x4_f32 (full f32 precision — the
// kernel is HBM-bandwidth bound on the weights, so low-precision WMMA buys
// nothing). Each wave computes a 32x16 partial tile (2 accumulators sharing
// the B registers), double-buffering K=16 stages (12 b64 loads / stage) so
// ~24 loads stay in flight per wave instead of 3. Partials are reduced
// through LDS in a fixed order (deterministic), bias added, then stored.
//
// A layout per ISA: lane l holds A[M=l&15][K=2*(l>>4)+v] for VGPR v -> one
// b64 (v2f) load per operand per WMMA. B (4x16, KxN) mirrors it with N=l&15.
// ---------------------------------------------------------------------------
__global__ __launch_bounds__(32*GEMM_WAVES) void rnnt_gemm_xwT(
    const float* __restrict__ X,     // (32, Kin)
    const float* __restrict__ W,     // (Fdim, Kin)
    const float* __restrict__ bias,  // (Fdim,)
    float* __restrict__ Y,           // (32, Fdim)
    int Kin, int Fdim)
{
    const int f0     = blockIdx.x * 16;
    const int tid    = threadIdx.x;
    const int w      = tid >> 5;     // wave id 0..7
    const int l      = tid & 31;     // lane
    const int half   = l >> 4;       // 0 or 1
    const int lane15 = l & 15;

    const int Kc = Kin / GEMM_WAVES; // multiple of 32 for both GEMMs
    const int kb = w * Kc;

    v8f c0 = {0.f,0.f,0.f,0.f,0.f,0.f,0.f,0.f};
    v8f c1 = {0.f,0.f,0.f,0.f,0.f,0.f,0.f,0.f};

    const float* xa0 = X + lane15 * Kin        + 2 * half + kb;  // batch rows 0..15
    const float* xa1 = X + (16 + lane15) * Kin + 2 * half + kb;  // batch rows 16..31
    const float* wb  = W + (f0 + lane15) * Kin + 2 * half + kb;  // W rows = F cols

    // Double-buffered operand registers: [buf][K-substep], stage = K advance 16
    v2f A0[2][4], A1[2][4], Bv[2][4];

#define LOAD_STAGE(BUF, KOFF)                                        \
    {                                                                \
        _Pragma("unroll")                                            \
        for (int i = 0; i < 4; ++i) {                                \
            A0[BUF][i] = *(const v2f*)(xa0 + (KOFF) + 4 * i);        \
            A1[BUF][i] = *(const v2f*)(xa1 + (KOFF) + 4 * i);        \
            Bv[BUF][i] = *(const v2f*)(wb  + (KOFF) + 4 * i);        \
        }                                                            \
    }

#define COMPUTE_STAGE(BUF)                                                     \
    {                                                                          \
        _Pragma("unroll")                                                      \
        for (int i = 0; i < 4; ++i) {                                          \
            c0 = __builtin_amdgcn_wmma_f32_16x16x4_f32(                        \
                     false, A0[BUF][i], false, Bv[BUF][i], (short)0, c0,       \
                     false, false);                                            \
            c1 = __builtin_amdgcn_wmma_f32_16x16x4_f32(                        \
                     false, A1[BUF][i], false, Bv[BUF][i], (short)0, c1,       \
                     false, false);                                            \
        }                                                                      \
    }

    // S = Kc/16 stages, S even. Prologue loads stage 0; main loop processes
    // two stages per iteration with compile-time buffer indices; final stage
    // peeled so we never prefetch past the chunk (no OOB read on last W row).
    LOAD_STAGE(0, 0)
    int k = 16;
    for (; k < Kc - 16; k += 32) {
        LOAD_STAGE(1, k)
        COMPUTE_STAGE(0)
        LOAD_STAGE(0, k + 16)
        COMPUTE_STAGE(1)
    }
    LOAD_STAGE(1, k)        // k == Kc - 16 here
    COMPUTE_STAGE(0)
    COMPUTE_STAGE(1)

#undef LOAD_STAGE
#undef COMPUTE_STAGE

    // Cross-wave reduction of the 8 partial 32x16 tiles (fixed order).
    __shared__ float racc[GEMM_WAVES][32][16];
#pragma unroll
    for (int r = 0; r < 8; ++r) {
        racc[w][l][r]     = c0[r];
        racc[w][l][8 + r] = c1[r];
    }
    __syncthreads();

    if (w == 0) {
        const float bf = bias[f0 + lane15];
        // C/D layout: VGPR r, lanes 0-15 -> M=r ; lanes 16-31 -> M=8+r ; N=l&15
#pragma unroll
        for (int r = 0; r < 8; ++r) {
            float s0 = 0.f, s1 = 0.f;
#pragma unroll
            for (int v = 0; v < GEMM_WAVES; ++v) {
                s0 += racc[v][l][r];
                s1 += racc[v][l][8 + r];
            }
            Y[(r + 8 * half) * Fdim      + f0 + lane15] = s0 + bf;
            Y[(16 + r + 8 * half) * Fdim + f0 + lane15] = s1 + bf;
        }
    }
}

// ---------------------------------------------------------------------------
// Kernel 2: per-(n,t,u) row: y_c = tanh(j0*W0[c] + j1*W1[c] + b[c]),
// lse = log(sum_c exp(y_c))  (y in [-1,1] so no max-shift needed),
// bl = y_0 - lse, lab = y_{ys[n,u]} - lse (masked by ylen, last col = NEG).
// One 256-thread block per (n,t); W_fc2/b_fc2 staged in LDS (48KB).
// ---------------------------------------------------------------------------
__global__ __launch_bounds__(256) void rnnt_row_lse(
    const float* __restrict__ enc,   // (32, F_)
    const float* __restrict__ pred,  // (32, F_)
    const float* __restrict__ Wfc,   // (C_, 2)
    const float* __restrict__ bfc,   // (C_,)
    const int*   __restrict__ ys,    // (32, U_-1)
    const int*   __restrict__ ylen,  // (32,)
    float* __restrict__ bl,          // (32, T_, U_)
    float* __restrict__ lab)         // (32, T_, U_)
{
    __shared__ float sw0[C_];
    __shared__ float sw1[C_];
    __shared__ float sb[C_];
    __shared__ float red[256];
    __shared__ float sh_y0, sh_ylab;

    const int nt  = blockIdx.x;
    const int n   = nt / T_;
    const int t   = nt % T_;
    const int tid = threadIdx.x;

    for (int c = tid; c < C_; c += 256) {
        float2 w = ((const float2*)Wfc)[c];
        sw0[c] = w.x; sw1[c] = w.y; sb[c] = bfc[c];
    }
    __syncthreads();

    const int yl = ylen[n];

    for (int u = 0; u < U_; ++u) {
        const int f  = (t * U_ + u) * 2;
        const float j0 = enc[n * F_ + f]     + pred[n * F_ + f];
        const float j1 = enc[n * F_ + f + 1] + pred[n * F_ + f + 1];
        const int yc = (u < U_ - 1) ? ys[n * (U_ - 1) + u] : -1;

        float s = 0.f;
        for (int c = tid; c < C_; c += 256) {
            float x = fmaf(j0, sw0[c], fmaf(j1, sw1[c], sb[c]));
            x = fminf(fmaxf(x, -15.f), 15.f);
            float e2 = __expf(2.f * x);
            float y  = (e2 - 1.f) / (e2 + 1.f);   // tanh(x)
            s += __expf(y);
            if (c == 0)  sh_y0   = y;
            if (c == yc) sh_ylab = y;
        }
        red[tid] = s;
        __syncthreads();
#pragma unroll
        for (int off = 128; off > 0; off >>= 1) {
            if (tid < off) red[tid] += red[tid + off];
            __syncthreads();
        }
        if (tid == 0) {
            const float lse = __logf(red[0]);
            const int idx = (n * T_ + t) * U_ + u;
            bl[idx] = sh_y0 - lse;
            float lv = NEGC;
            if (u < U_ - 1 && u < yl) lv = sh_ylab - lse;
            lab[idx] = lv;
        }
        __syncthreads();  // protect red/sh_* for next u
    }
}

// ---------------------------------------------------------------------------
// Kernel 3: transducer forward DP. One lane per batch element (1 wave total).
// alpha recurrence is a serial scan in u, so lane-per-n is the natural shape.
// ---------------------------------------------------------------------------
__device__ __forceinline__ float logaddexp_f(float a, float b) {
    float m = fmaxf(a, b);
    float d = fabsf(a - b);
    return m + log1pf(__expf(-d));
}

__global__ __launch_bounds__(32) void rnnt_dp(
    const float* __restrict__ bl,    // (32, T_, U_)
    const float* __restrict__ lab,   // (32, T_, U_)
    const int*   __restrict__ ylen,  // (32,)
    float* __restrict__ out)         // (1,)
{
    const int n = threadIdx.x;
    const float* BL = bl  + n * T_ * U_;
    const float* LB = lab + n * T_ * U_;

    float alpha[U_];
    alpha[0] = 0.f;
#pragma unroll
    for (int u = 1; u < U_; ++u) alpha[u] = alpha[u - 1] + LB[u - 1];

    for (int t = 1; t < T_; ++t) {
        float c = NEGC;
#pragma unroll
        for (int u = 0; u < U_; ++u) {
            float a = alpha[u] + BL[(t - 1) * U_ + u];
            float lv = (u == 0) ? NEGC : LB[t * U_ + (u - 1)];
            float v = logaddexp_f(a, c + lv);
            alpha[u] = v;
            c = v;
        }
    }

    const int yl = ylen[n];
    float av = 0.f;
#pragma unroll
    for (int u = 0; u < U_; ++u) if (u == yl) av = alpha[u];
    const float ll = av + BL[(T_ - 1) * U_ + yl];

    __shared__ float sll[32];
    sll[n] = ll;
    __syncthreads();
    if (n == 0) {
        float s = 0.f;
        for (int i = 0; i < 32; ++i) s += sll[i];
        out[0] = -s / (float)N_;
    }
}

// ---------------------------------------------------------------------------
extern "C" void kernel_launch(void* const* d_in, const int* in_sizes, int n_in,
                              void* d_out, int out_size, void* d_ws, size_t ws_size,
                              hipStream_t stream) {
    (void)in_sizes; (void)n_in; (void)out_size; (void)ws_size;

    const float* encoder_output    = (const float*)d_in[0];  // (32,76,512)
    const float* prediction_output = (const float*)d_in[1];  // (32,31,512)
    // d_in[2] xs, d_in[3] xlen: unused by forward
    const int*   ys    = (const int*)d_in[4];                // (32,19)
    const int*   ylen  = (const int*)d_in[5];                // (32,)
    const float* W_enc  = (const float*)d_in[6];             // (3040,38912)
    const float* b_enc  = (const float*)d_in[7];
    const float* W_pred = (const float*)d_in[8];             // (3040,15872)
    const float* b_pred = (const float*)d_in[9];
    const float* W_fc2  = (const float*)d_in[10];            // (4096,2)
    const float* b_fc2  = (const float*)d_in[11];

    float* ws   = (float*)d_ws;
    float* enc  = ws;                  // 32*3040
    float* pred = enc + N_ * F_;       // 32*3040
    float* bl   = pred + N_ * F_;      // 32*76*20
    float* lab  = bl + N_ * T_ * U_;   // 32*76*20

    // GEMMs: 190 F-tiles x 8 K-split waves = 1520 waves each
    rnnt_gemm_xwT<<<F_ / 16, 32 * GEMM_WAVES, 0, stream>>>(
        encoder_output, W_enc, b_enc, enc, ENC_IN_, F_);
    rnnt_gemm_xwT<<<F_ / 16, 32 * GEMM_WAVES, 0, stream>>>(
        prediction_output, W_pred, b_pred, pred, PRED_IN_, F_);

    // Row-wise tanh+LSE: one block per (n,t)
    rnnt_row_lse<<<N_ * T_, 256, 0, stream>>>(
        enc, pred, W_fc2, b_fc2, ys, ylen, bl, lab);

    // Forward DP + mean
    rnnt_dp<<<1, 32, 0, stream>>>(bl, lab, ylen, (float*)d_out);
}